// SinkhornDistance_27693949125013
// MI455X (gfx1250) — compile-verified
//
#include <hip/hip_runtime.h>
#include <hip/hip_bf16.h>
#include <cstddef>

typedef float v2f __attribute__((ext_vector_type(2)));
typedef float v8f __attribute__((ext_vector_type(8)));

#define NB 8
#define P 1024
#define DIMF 64

__constant__ constexpr float EPS_F    = 0.1f;
__constant__ constexpr float INV_EPS  = 10.0f;
// log(1/1024 + 1e-8)
__constant__ constexpr float LOG_MU   = -6.9314616f;
__constant__ constexpr float LOG_NU   = -6.9314616f;
__constant__ constexpr float THRESH_N = 0.8f;   // THRESH * N  (err is mean over N of row-sum)

// ---------------- init: u=v=0, done=0 ----------------
__global__ void sk_init(float* __restrict__ u, float* __restrict__ v, int* __restrict__ done) {
    int i = blockIdx.x * blockDim.x + threadIdx.x;
    if (i < NB * P) { u[i] = 0.0f; v[i] = 0.0f; }
    if (i == 0) *done = 0;
}

// ---------------- row norms ||x_i||^2, ||y_j||^2 ----------------
__global__ void sk_norms(const float* __restrict__ x, const float* __restrict__ y,
                         float* __restrict__ x2, float* __restrict__ y2) {
    int r = blockIdx.x * blockDim.x + threadIdx.x;   // 0..8191
    if (r >= NB * P) return;
    const float4* xr = (const float4*)(x + (size_t)r * DIMF);
    const float4* yr = (const float4*)(y + (size_t)r * DIMF);
    float sx = 0.0f, sy = 0.0f;
#pragma unroll
    for (int k = 0; k < DIMF / 4; ++k) {
        float4 a = xr[k]; sx += a.x * a.x + a.y * a.y + a.z * a.z + a.w * a.w;
        float4 b = yr[k]; sy += b.x * b.x + b.y * b.y + b.z * b.z + b.w * b.w;
    }
    x2[r] = sx; y2[r] = sy;
}

// ---------------- cost matrix via fp32 WMMA 16x16x4 ----------------
// One wave computes a 16x64 strip of C for one batch (4 accumulators share A).
// fp32 A layout: lanes 0-15 hold M=lane, K={k0,k0+1}; lanes 16-31 hold M=lane-16, K={k0+2,k0+3}.
// B (4x16) dual layout: lane holds N=lane&15, K pair selected by lane>=16.
// Each lane's accumulator column (8 consecutive rows, one col) also gives a free
// register-level transpose: two aligned float4 stores per tile produce C^T.
__global__ void sk_cost(const float* __restrict__ x, const float* __restrict__ y,
                        const float* __restrict__ x2, const float* __restrict__ y2,
                        float* __restrict__ C, float* __restrict__ Ct) {
    int wave = threadIdx.x >> 5;
    int lane = threadIdx.x & 31;
    int strip = blockIdx.x * (blockDim.x >> 5) + wave;   // 0..8191
    int nb  = strip >> 10;        // 1024 strips per batch
    int rem = strip & 1023;
    int ti  = rem >> 4;           // 64 row tiles of 16
    int tj  = rem & 15;           // 16 col strips of 64
    int half = lane >> 4;
    int l15  = lane & 15;

    const float* arow = x + ((size_t)nb * P + (size_t)(ti * 16 + l15)) * DIMF + half * 2;
    const float* brow = y + ((size_t)nb * P + (size_t)(tj * 64 + l15)) * DIMF + half * 2;

    v8f acc[4] = {};
#pragma unroll
    for (int k0 = 0; k0 < DIMF; k0 += 4) {
        v2f a  = *(const v2f*)(arow + k0);
        v2f b0 = *(const v2f*)(brow + k0);
        v2f b1 = *(const v2f*)(brow + 16 * DIMF + k0);
        v2f b2 = *(const v2f*)(brow + 32 * DIMF + k0);
        v2f b3 = *(const v2f*)(brow + 48 * DIMF + k0);
        acc[0] = __builtin_amdgcn_wmma_f32_16x16x4_f32(false, a, false, b0, (short)0, acc[0], false, false);
        acc[1] = __builtin_amdgcn_wmma_f32_16x16x4_f32(false, a, false, b1, (short)0, acc[1], false, false);
        acc[2] = __builtin_amdgcn_wmma_f32_16x16x4_f32(false, a, false, b2, (short)0, acc[2], false, false);
        acc[3] = __builtin_amdgcn_wmma_f32_16x16x4_f32(false, a, false, b3, (short)0, acc[3], false, false);
    }

    float* Cb  = C + (size_t)nb * P * P;
    float* Ctb = Ct ? (Ct + (size_t)nb * P * P) : nullptr;
    int rowbase = ti * 16 + half * 8;   // C/D layout: VGPR r -> M = r + 8*(lane>=16)
#pragma unroll
    for (int jt = 0; jt < 4; ++jt) {
        int col = tj * 64 + jt * 16 + l15;
        float yc = y2[nb * P + col];
        float vals[8];
#pragma unroll
        for (int r = 0; r < 8; ++r) {
            int row = rowbase + r;
            float val = fmaxf(x2[nb * P + row] + yc - 2.0f * acc[jt][r], 0.0f);
            vals[r] = val;
            Cb[(size_t)row * P + col] = val;
        }
        if (Ctb) {
            float4* dst = (float4*)(Ctb + (size_t)col * P + rowbase);  // 32B aligned
            dst[0] = make_float4(vals[0], vals[1], vals[2], vals[3]);
            dst[1] = make_float4(vals[4], vals[5], vals[6], vals[7]);
        }
    }
}

// ---------------- u update: per-row logsumexp ----------------
// u_new_i = EPS*(LOG_MU - LSE_j((v_j - C_ij)/EPS))   (independent of old u)
__global__ void sk_uupd(const float* __restrict__ C, const float* __restrict__ v,
                        float* __restrict__ u, float* __restrict__ udiff,
                        const int* __restrict__ done) {
    if (*done) return;
    __shared__ float red[256];
    int row = blockIdx.x;              // 0..8191 flat (nb,i)
    int t = threadIdx.x;
    const float* crow = C + (size_t)row * P;
    const float* vb = v + ((row >> 10) << 10);
    float tv[4];
    float m = -3.4e38f;
#pragma unroll
    for (int s = 0; s < 4; ++s) {
        int j = t + s * 256;
        tv[s] = (vb[j] - crow[j]) * INV_EPS;
        m = fmaxf(m, tv[s]);
    }
    red[t] = m; __syncthreads();
    for (int off = 128; off > 0; off >>= 1) {
        if (t < off) red[t] = fmaxf(red[t], red[t + off]);
        __syncthreads();
    }
    m = red[0]; __syncthreads();
    float s0 = 0.0f;
#pragma unroll
    for (int s = 0; s < 4; ++s) s0 += __expf(tv[s] - m);
    red[t] = s0; __syncthreads();
    for (int off = 128; off > 0; off >>= 1) {
        if (t < off) red[t] += red[t + off];
        __syncthreads();
    }
    if (t == 0) {
        float L = m + __logf(red[0]);
        float un = EPS_F * (LOG_MU - L);
        udiff[row] = fabsf(un - u[row]);
        u[row] = un;
    }
}

// ---------------- v update (transposed path): mirror of u update on C^T ----------------
__global__ void sk_vupd_t(const float* __restrict__ Ct, const float* __restrict__ u,
                          float* __restrict__ v, const int* __restrict__ done) {
    if (*done) return;
    __shared__ float red[256];
    int col = blockIdx.x;              // 0..8191 flat (nb,j)
    int t = threadIdx.x;
    const float* crow = Ct + (size_t)col * P;
    const float* ub = u + ((col >> 10) << 10);
    float tv[4];
    float m = -3.4e38f;
#pragma unroll
    for (int s = 0; s < 4; ++s) {
        int i = t + s * 256;
        tv[s] = (ub[i] - crow[i]) * INV_EPS;
        m = fmaxf(m, tv[s]);
    }
    red[t] = m; __syncthreads();
    for (int off = 128; off > 0; off >>= 1) {
        if (t < off) red[t] = fmaxf(red[t], red[t + off]);
        __syncthreads();
    }
    m = red[0]; __syncthreads();
    float s0 = 0.0f;
#pragma unroll
    for (int s = 0; s < 4; ++s) s0 += __expf(tv[s] - m);
    red[t] = s0; __syncthreads();
    for (int off = 128; off > 0; off >>= 1) {
        if (t < off) red[t] += red[t + off];
        __syncthreads();
    }
    if (t == 0) {
        float L = m + __logf(red[0]);
        v[col] = EPS_F * (LOG_NU - L);
    }
}

// ---------------- v update (fallback, column-wise on C, two passes) ----------------
__global__ void sk_vupd_col(const float* __restrict__ C, const float* __restrict__ u,
                            float* __restrict__ v, const int* __restrict__ done) {
    if (*done) return;
    __shared__ float smax[256];
    __shared__ float colred[64];
    int t = threadIdx.x;
    int tx = t & 63, ty = t >> 6;
    int nb = blockIdx.x >> 4;
    int jb = (blockIdx.x & 15) * 64;
    int j = jb + tx;
    const float* Cb = C + (size_t)nb * P * P;
    const float* ub = u + nb * P;

    float m = -3.4e38f;
    for (int i = ty; i < P; i += 4)
        m = fmaxf(m, (ub[i] - Cb[(size_t)i * P + j]) * INV_EPS);
    smax[t] = m; __syncthreads();
    if (ty == 0)
        colred[tx] = fmaxf(fmaxf(smax[tx], smax[tx + 64]),
                           fmaxf(smax[tx + 128], smax[tx + 192]));
    __syncthreads();
    m = colred[tx];
    float s = 0.0f;
    for (int i = ty; i < P; i += 4)
        s += __expf((ub[i] - Cb[(size_t)i * P + j]) * INV_EPS - m);
    smax[t] = s; __syncthreads();
    if (ty == 0) {
        float tot = smax[tx] + smax[tx + 64] + smax[tx + 128] + smax[tx + 192];
        v[nb * P + j] = EPS_F * (LOG_NU - (m + __logf(tot)));
    }
}

// ---------------- convergence check (deterministic tree) ----------------
__global__ void sk_check(const float* __restrict__ udiff, int* __restrict__ done) {
    if (*done) return;
    __shared__ float red[256];
    int t = threadIdx.x;
    float s = 0.0f;
#pragma unroll
    for (int k = 0; k < 32; ++k) s += udiff[t + k * 256];
    red[t] = s; __syncthreads();
    for (int off = 128; off > 0; off >>= 1) {
        if (t < off) red[t] += red[t + off];
        __syncthreads();
    }
    if (t == 0 && red[0] < THRESH_N) *done = 1;
}

// ---------------- final cost: sum exp((u+v-C)/eps) * C ----------------
__global__ void sk_final(const float* __restrict__ C, const float* __restrict__ u,
                         const float* __restrict__ v, float* __restrict__ partials) {
    __shared__ float red[256];
    int t = threadIdx.x;
    int nb = blockIdx.x >> 5;
    int rb = (blockIdx.x & 31) * 32;
    const float* Cb = C + (size_t)nb * P * P;
    const float* ub = u + nb * P;
    const float* vb = v + nb * P;
    float acc = 0.0f;
    for (int idx = t; idx < 32 * P; idx += 256) {
        int i = rb + (idx >> 10);
        int j = idx & 1023;
        float c = Cb[(size_t)i * P + j];
        acc += __expf((ub[i] + vb[j] - c) * INV_EPS) * c;
    }
    red[t] = acc; __syncthreads();
    for (int off = 128; off > 0; off >>= 1) {
        if (t < off) red[t] += red[t + off];
        __syncthreads();
    }
    if (t == 0) partials[blockIdx.x] = red[0];
}

__global__ void sk_out(const float* __restrict__ partials, float* __restrict__ out) {
    __shared__ float red[32];
    int t = threadIdx.x;
    red[t] = partials[blockIdx.x * 32 + t];
    __syncthreads();
    for (int off = 16; off > 0; off >>= 1) {
        if (t < off) red[t] += red[t + off];
        __syncthreads();
    }
    if (t == 0) out[blockIdx.x] = red[0];
}

extern "C" void kernel_launch(void* const* d_in, const int* in_sizes, int n_in,
                              void* d_out, int out_size, void* d_ws, size_t ws_size,
                              hipStream_t stream) {
    (void)in_sizes; (void)n_in; (void)out_size;
    const float* x = (const float*)d_in[0];
    const float* y = (const float*)d_in[1];
    float* out = (float*)d_out;

    const size_t matElems  = (size_t)NB * P * P;          // 8M floats = 32 MB
    const size_t smallElems = 5 * (size_t)NB * P + 256 + 64;
    const bool useT = ws_size >= (2 * matElems + smallElems) * sizeof(float);

    float* C  = (float*)d_ws;
    float* Ct = useT ? (C + matElems) : nullptr;
    float* base     = useT ? (Ct + matElems) : (C + matElems);
    float* x2       = base;                 // 8192
    float* y2       = x2 + NB * P;          // 8192
    float* u        = y2 + NB * P;          // 8192
    float* v        = u  + NB * P;          // 8192
    float* udiff    = v  + NB * P;          // 8192
    float* partials = udiff + NB * P;       // 256
    int*   done     = (int*)(partials + 256);

    sk_init <<<32,   256, 0, stream>>>(u, v, done);
    sk_norms<<<32,   256, 0, stream>>>(x, y, x2, y2);
    sk_cost <<<2048, 128, 0, stream>>>(x, y, x2, y2, C, Ct);  // 8192 waves, 16x64 strip each

    for (int it = 0; it < 100; ++it) {
        sk_uupd<<<NB * P, 256, 0, stream>>>(C, v, u, udiff, done);
        if (useT)
            sk_vupd_t<<<NB * P, 256, 0, stream>>>(Ct, u, v, done);
        else
            sk_vupd_col<<<NB * 16, 256, 0, stream>>>(C, u, v, done);
        sk_check<<<1, 256, 0, stream>>>(udiff, done);
    }

    sk_final<<<NB * 32, 256, 0, stream>>>(C, u, v, partials);
    sk_out  <<<NB,      32,  0, stream>>>(partials, out);
}